// QPTHSolver_15118284882650
// MI455X (gfx1250) — compile-verified
//
#include <hip/hip_runtime.h>
#include <hip/hip_bf16.h>

// ---------------------------------------------------------------------------
// Batched IPM QP solver for MI455X (gfx1250).
// One workgroup (256 threads = 8 wave32) per batch element; the whole
// 208x208 padded KKT matrix lives in LDS (173KB of the 320KB/WGP pool).
// Trailing rank-16 LU updates use V_WMMA_F32_16X16X4_F32 with A-fragment
// reuse across paired output tiles. Inputs staged via async global->LDS.
// ---------------------------------------------------------------------------

#define NDIM   14
#define NV     196           // n = N*N
#define MC     224           // m = n + 2N
#define NP     208           // padded KKT dim (13 * 16)
#define LDM    208           // row stride of KKT matrix in LDS
#define NBLK   13            // 16x16 tile blocks per side
#define L2REG  0.0001f
#define SIGMA  0.1f
#define FTB    0.99f
#define NITER  25

typedef __attribute__((ext_vector_type(2))) float v2f;
typedef __attribute__((ext_vector_type(4))) float v4f;
typedef __attribute__((ext_vector_type(8))) float v8f;

// LDS layout (floats):
//   Mat   : NP*LDM            (43264)
//   xv,sv,zv,pv,hv,dv,rzv,cv,rhs,gdx : 10 * 224
//   red   : 256
//   scal  : 16   [0]=y [1]=beq [2]=mu [3]=alpha [4]=tmp/ry [5]=tmp
#define PV_OFF     (NP*LDM + 3*224)          // float index of pv[] in LDS
#define LDS_FLOATS (NP*LDM + 10*224 + 256 + 16)

__global__ __launch_bounds__(256)
void qp_ipm_kernel(const float* __restrict__ dm,
                   const float* __restrict__ w1,
                   const float* __restrict__ w2,
                   float* __restrict__ out)
{
    extern __shared__ float lds[];
    float* Mat = lds;
    float* xv  = lds + NP*LDM;
    float* sv  = xv  + 224;
    float* zv  = sv  + 224;
    float* pv  = zv  + 224;      // == lds + PV_OFF
    float* hv  = pv  + 224;
    float* dv  = hv  + 224;
    float* rzv = dv  + 224;
    float* cv  = rzv + 224;
    float* rhs = cv  + 224;
    float* gdx = rhs + 224;
    float* red = gdx + 224;
    float* scal = red + 256;

    const int b   = blockIdx.x;
    const int tid = threadIdx.x;
    const int wave  = tid >> 5;
    const int lane  = tid & 31;
    const int lhalf = lane >> 4;     // 0: lanes 0-15, 1: lanes 16-31
    const int l15   = lane & 15;

    // ---- stage p = dm into LDS via CDNA5 async global->LDS ----------------
#if defined(__gfx1250__)
    if (tid < NV) {
        unsigned ldsoff = (unsigned)((PV_OFF + tid) * 4);
        unsigned voff   = (unsigned)(tid * 4);
        const float* gbase = dm + (size_t)b * NV;
        asm volatile("global_load_async_to_lds_b32 %0, %1, %2"
                     :: "v"(ldsoff), "v"(voff), "s"(gbase)
                     : "memory");
    }
    asm volatile("s_wait_asynccnt 0" ::: "memory");
#else
    if (tid < NV) pv[tid] = dm[(size_t)b * NV + tid];
#endif

    // ---- setup: normalize weights, build h / beq, init state --------------
    if (tid < 28) {
        red[tid] = (tid < NDIM) ? w1[b*NDIM + tid] : w2[b*NDIM + tid - NDIM];
    }
    __syncthreads();
    if (tid == 0) {
        float s1 = 0.f, s2 = 0.f;
        for (int i = 0; i < NDIM; ++i) { s1 += red[i]; s2 += red[NDIM + i]; }
        float sc1 = (float)NDIM / s1, sc2 = (float)NDIM / s2;
        scal[4] = sc1; scal[5] = sc2;
        scal[1] = fminf(s1 * sc1, s2 * sc2);   // beq (== N)
        scal[0] = 0.f;                          // y
    }
    __syncthreads();
    const float beq = scal[1];
    for (int i = tid; i < MC; i += 256) {
        float hval = 0.f;
        if (i >= NV && i < NV + NDIM)      hval = red[i - NV] * scal[4];
        else if (i >= NV + NDIM)           hval = red[i - NV] * scal[5];
        hv[i] = hval;
        sv[i] = 1.f;
        zv[i] = 1.f;
        if (i < NV) xv[i] = beq / (float)NV;
    }
    __syncthreads();

    // ---- IPM iterations ----------------------------------------------------
    for (int it = 0; it < NITER; ++it) {
        // mu = mean(s*z)
        {
            float part = 0.f;
            for (int i = tid; i < MC; i += 256) part += sv[i] * zv[i];
            red[tid] = part;
            __syncthreads();
            if (tid == 0) {
                float t = 0.f;
                for (int i = 0; i < 256; ++i) t += red[i];
                scal[2] = t / (float)MC;
            }
            __syncthreads();
        }
        const float mu = scal[2];

        // d, rz, c  (x @ G^T via row/col sums — G is Kronecker-structured)
        for (int j = tid; j < MC; j += 256) {
            float gx;
            if (j < NV) gx = -xv[j];
            else if (j < NV + NDIM) {
                int r = j - NV; float t = 0.f;
                for (int c0 = 0; c0 < NDIM; ++c0) t += xv[r*NDIM + c0];
                gx = t;
            } else {
                int c0 = j - NV - NDIM; float t = 0.f;
                for (int r = 0; r < NDIM; ++r) t += xv[r*NDIM + c0];
                gx = t;
            }
            float rzj = gx + sv[j] - hv[j];
            float dj  = zv[j] / sv[j];
            dv[j]  = dj;
            rzv[j] = rzj;
            cv[j]  = (zv[j]*rzj - sv[j]*zv[j] + SIGMA*mu) / sv[j];
        }
        // ry = sum(x) - beq
        {
            float part = 0.f;
            for (int i = tid; i < NV; i += 256) part += xv[i];
            red[tid] = part;
            __syncthreads();
            if (tid == 0) {
                float t = 0.f;
                for (int i = 0; i < 256; ++i) t += red[i];
                scal[4] = t - beq;
            }
            __syncthreads();
        }
        // rhs = [-(rx + c@G), -ry, 0-pad]
        for (int i = tid; i < NP; i += 256) {
            if (i < NV) {
                int r = i / NDIM, c0 = i % NDIM;
                float zg = -zv[i] + zv[NV + r] + zv[NV + NDIM + c0];
                float cg = -cv[i] + cv[NV + r] + cv[NV + NDIM + c0];
                float rx = L2REG*xv[i] + pv[i] + zg + scal[0];
                rhs[i] = -(rx + cg);
            } else if (i == NV) rhs[i] = -scal[4];
            else rhs[i] = 0.f;
        }
        // Build KKT matrix M (structured: no GEMM for G^T D G), b128 stores
        for (int idx4 = tid; idx4 < (NP*NP)/4; idx4 += 256) {
            const int base = idx4 * 4;
            const int i  = base / NP;          // LDM % 4 == 0: one row per v4f
            const int j0 = base - i * NP;
            v4f v;
            #pragma unroll
            for (int q = 0; q < 4; ++q) {
                const int j = j0 + q;
                float e;
                if (i < NV && j < NV) {
                    int ri = i / NDIM, ci = i % NDIM;
                    int rj = j / NDIM, cj = j % NDIM;
                    e = 0.f;
                    if (i == j)   e += L2REG + dv[i];
                    if (ri == rj) e += dv[NV + ri];
                    if (ci == cj) e += dv[NV + NDIM + ci];
                } else if (i == NV && j < NV)  e = 1.f;
                else if (j == NV && i < NV)    e = 1.f;
                else if (i == NV && j == NV)   e = 0.f;
                else                           e = (i == j) ? 1.f : 0.f;
                v[q] = e;
            }
            *(v4f*)&Mat[i*LDM + j0] = v;
        }
        __syncthreads();

        // ---- blocked LU (no pivoting), WMMA trailing updates --------------
        for (int kb = 0; kb < NBLK; ++kb) {
            const int k0 = kb * 16;
            // panel factorization (16 sequential column steps)
            for (int kk = 0; kk < 16; ++kk) {
                const int col = k0 + kk;
                const float ipiv = 1.0f / Mat[col*LDM + col];
                for (int r = col + 1 + tid; r < NP; r += 256) {
                    float l = Mat[r*LDM + col] * ipiv;
                    Mat[r*LDM + col] = l;
                    for (int cc = kk + 1; cc < 16; ++cc)
                        Mat[r*LDM + k0 + cc] -= l * Mat[col*LDM + k0 + cc];
                }
                __syncthreads();
            }
            // row-panel triangular solve: U[kb][j>kb] = L_kk^{-1} A[kb][j]
            for (int j = k0 + 16 + tid; j < NP; j += 256) {
                for (int kk = 1; kk < 16; ++kk) {
                    float acc = Mat[(k0 + kk)*LDM + j];
                    for (int t0 = 0; t0 < kk; ++t0)
                        acc -= Mat[(k0 + kk)*LDM + k0 + t0] * Mat[(k0 + t0)*LDM + j];
                    Mat[(k0 + kk)*LDM + j] = acc;
                }
            }
            __syncthreads();
            // trailing update: A[i][j] -= L[i][kb] * U[kb][j]  via f32 WMMA.
            // Each wave owns a 16x32 strip: A fragment loaded (negated) once,
            // reused across two j-tiles => 8 WMMAs per A load.
            const int nt = NBLK - 1 - kb;
            const int npair = (nt + 1) >> 1;
            auto tile_update = [&](int r0, int c0, const v2f* a) {
                v8f acc;
                #pragma unroll
                for (int g = 0; g < 8; ++g)     // C/D: VGPR g = rows g / g+8
                    acc[g] = Mat[(r0 + g + lhalf*8)*LDM + c0 + l15];
                #pragma unroll
                for (int ks = 0; ks < 4; ++ks) {
                    const int ak = k0 + ks*4 + lhalf*2;
                    v2f bmat;
                    bmat[0] = Mat[ak*LDM + c0 + l15];       // B 4x16 layout
                    bmat[1] = Mat[(ak + 1)*LDM + c0 + l15];
                    acc = __builtin_amdgcn_wmma_f32_16x16x4_f32(
                        false, a[ks], false, bmat, (short)0, acc, false, false);
                }
                #pragma unroll
                for (int g = 0; g < 8; ++g)
                    Mat[(r0 + g + lhalf*8)*LDM + c0 + l15] = acc[g];
            };
            for (int t = wave; t < nt*npair; t += 8) {
                const int ib = kb + 1 + t / npair;
                const int jp = (t % npair) * 2;
                const int r0 = ib * 16;
                v2f a[4];                        // negated A 16x16, 4 k-slices
                #pragma unroll
                for (int ks = 0; ks < 4; ++ks) {
                    const int ak = k0 + ks*4 + lhalf*2;   // A 16x4 layout
                    a[ks][0] = -Mat[(r0 + l15)*LDM + ak];
                    a[ks][1] = -Mat[(r0 + l15)*LDM + ak + 1];
                }
                tile_update(r0, (kb + 1 + jp) * 16, a);
                if (jp + 1 < nt)                 // wave-uniform branch
                    tile_update(r0, (kb + 2 + jp) * 16, a);
            }
            __syncthreads();
        }

        // ---- forward substitution (unit lower) ----------------------------
        for (int kb = 0; kb < NBLK; ++kb) {
            const int k0 = kb * 16;
            if (tid == 0) {
                for (int kk = 0; kk < 16; ++kk) {
                    float v = rhs[k0 + kk];
                    for (int t0 = 0; t0 < kk; ++t0)
                        v -= Mat[(k0 + kk)*LDM + k0 + t0] * rhs[k0 + t0];
                    rhs[k0 + kk] = v;
                }
            }
            __syncthreads();
            for (int r = k0 + 16 + tid; r < NP; r += 256) {
                float acc = 0.f;
                for (int t0 = 0; t0 < 16; ++t0)
                    acc += Mat[r*LDM + k0 + t0] * rhs[k0 + t0];
                rhs[r] -= acc;
            }
            __syncthreads();
        }
        // ---- backward substitution (upper, non-unit diagonal) -------------
        for (int kb = NBLK - 1; kb >= 0; --kb) {
            const int k0 = kb * 16;
            if (tid == 0) {
                for (int kk = 15; kk >= 0; --kk) {
                    float v = rhs[k0 + kk];
                    for (int t0 = kk + 1; t0 < 16; ++t0)
                        v -= Mat[(k0 + kk)*LDM + k0 + t0] * rhs[k0 + t0];
                    rhs[k0 + kk] = v / Mat[(k0 + kk)*LDM + k0 + kk];
                }
            }
            __syncthreads();
            for (int r = tid; r < k0; r += 256) {
                float acc = 0.f;
                for (int t0 = 0; t0 < 16; ++t0)
                    acc += Mat[r*LDM + k0 + t0] * rhs[k0 + t0];
                rhs[r] -= acc;
            }
            __syncthreads();
        }

        // ---- step: dx = rhs[0:196], dy = rhs[196] -------------------------
        for (int j = tid; j < MC; j += 256) {
            float g;
            if (j < NV) g = -rhs[j];
            else if (j < NV + NDIM) {
                int r = j - NV; float t = 0.f;
                for (int c0 = 0; c0 < NDIM; ++c0) t += rhs[r*NDIM + c0];
                g = t;
            } else {
                int c0 = j - NV - NDIM; float t = 0.f;
                for (int r = 0; r < NDIM; ++r) t += rhs[r*NDIM + c0];
                g = t;
            }
            gdx[j] = g;
        }
        __syncthreads();
        // alpha = min(1, FTB * min over fraction-to-boundary ratios)
        {
            float amin = 1e30f;
            for (int j = tid; j < MC; j += 256) {
                float dsv = -rzv[j] - gdx[j];
                float dzv = dv[j]*gdx[j] + cv[j];
                if (dsv < 0.f) amin = fminf(amin, -sv[j] / dsv);
                if (dzv < 0.f) amin = fminf(amin, -zv[j] / dzv);
            }
            red[tid] = amin;
            __syncthreads();
            if (tid == 0) {
                float t = 1e30f;
                for (int i = 0; i < 256; ++i) t = fminf(t, red[i]);
                scal[3] = fminf(1.0f, FTB * t);
            }
            __syncthreads();
        }
        const float alpha = scal[3];
        for (int j = tid; j < MC; j += 256) {
            float dsv = -rzv[j] - gdx[j];
            float dzv = dv[j]*gdx[j] + cv[j];
            sv[j] += alpha * dsv;
            zv[j] += alpha * dzv;
            if (j < NV) xv[j] += alpha * rhs[j];
        }
        if (tid == 0) scal[0] += alpha * rhs[NV];
        __syncthreads();
    }

    // ---- output flow = x ---------------------------------------------------
    for (int i = tid; i < NV; i += 256)
        out[b*NV + i] = xv[i];
}

extern "C" void kernel_launch(void* const* d_in, const int* in_sizes, int n_in,
                              void* d_out, int out_size, void* d_ws, size_t ws_size,
                              hipStream_t stream) {
    (void)n_in; (void)d_ws; (void)ws_size; (void)out_size;
    const float* dm = (const float*)d_in[0];   // (750,1,14,14) f32
    const float* w1 = (const float*)d_in[1];   // (750,1,14)    f32
    const float* w2 = (const float*)d_in[2];   // (750,1,14)    f32
    float* out = (float*)d_out;                // (750,14,14)   f32
    const int batch = in_sizes[0] / NV;
    const size_t lds_bytes = (size_t)LDS_FLOATS * sizeof(float);
    qp_ipm_kernel<<<batch, 256, lds_bytes, stream>>>(dm, w1, w2, out);
}